// TempGT_80582176408299
// MI455X (gfx1250) — compile-verified
//
#include <hip/hip_runtime.h>
#include <hip/hip_bf16.h>
#include <math.h>

typedef __attribute__((ext_vector_type(2))) float v2f;
typedef __attribute__((ext_vector_type(8))) float v8f;

#define B_SZ 1024
#define K_NB 32
#define G_NB 1024
#define NDIM 128
#define EDIM 128
#define TDIM 100
#define CDIM 228   // EDIM + TDIM
#define VLD  232   // padded row stride for v (even, 928B = 16B-aligned rows)

// ---------------------------------------------------------------------------
// bias_pooled[c] = (sum_k w_final[k]) * b_edge[c] + b_final[0]
// ---------------------------------------------------------------------------
__global__ void prep_bias_kernel(const float* __restrict__ w_final,
                                 const float* __restrict__ b_edge,
                                 const float* __restrict__ b_final,
                                 float* __restrict__ bias) {
  __shared__ float s;
  if (threadIdx.x == 0) {
    float acc = 0.f;
    for (int k = 0; k < K_NB; ++k) acc += w_final[k];
    s = acc;
  }
  __syncthreads();
  if (threadIdx.x < EDIM) bias[threadIdx.x] = b_edge[threadIdx.x] * s + b_final[0];
}

// ---------------------------------------------------------------------------
// Per batch row b:
//   v[b, 0:128]   = sum_k w_final[k] * edge_raw_features[eid[b,k], :]
//   v[b, 128:228] = sum_k w_final[k] * valid[b,k] * cos(dt[b,k]*time_w + time_b)
//   x2[b, 128:256] = masked-softmax-mean of node_raw_features over G time-gap
//                    neighbors + node_raw_features[node_ids[b]]
// All big gathers use float4 (global_load_b128): one row = one 512B wave txn.
// ---------------------------------------------------------------------------
__global__ __launch_bounds__(256)
void gather_kernel(const int* __restrict__ node_ids,
                   const float* __restrict__ t0,
                   const int* __restrict__ nbr_ids,
                   const int* __restrict__ nbr_eids,
                   const float* __restrict__ nbr_times,
                   const int* __restrict__ tg_ids,
                   const float* __restrict__ node_feat,
                   const float* __restrict__ edge_feat,
                   const float* __restrict__ time_w,
                   const float* __restrict__ time_b,
                   const float* __restrict__ w_final,
                   float* __restrict__ v,
                   float* __restrict__ x2) {
  const int b = blockIdx.x;
  const int t = threadIdx.x;

  __shared__ float  s_w[K_NB];
  __shared__ float  s_dt[K_NB];
  __shared__ int    s_valid[K_NB];
  __shared__ int    s_eid[K_NB];
  __shared__ int    s_gid[G_NB];
  __shared__ float4 s_red[256];
  __shared__ int    s_cnt;

  if (t == 0) s_cnt = 0;
  if (t < K_NB) {
    s_w[t]     = w_final[t];
    s_dt[t]    = t0[b] - nbr_times[b * K_NB + t];
    s_valid[t] = (nbr_ids[b * K_NB + t] != 0) ? 1 : 0;
    s_eid[t]   = nbr_eids[b * K_NB + t];
  }
  // stage time-gap neighbor ids + count valid ones
  int localcnt = 0;
  for (int i = t; i < G_NB; i += 256) {
    int id = tg_ids[b * G_NB + i];
    s_gid[i] = id;
    if (id > 0) localcnt++;
  }
  atomicAdd(&s_cnt, localcnt);
  __syncthreads();

  const int  cnt    = s_cnt;
  const bool useAll = (cnt == 0);
  const float scale = useAll ? (1.0f / ((float)G_NB * (float)G_NB))
                             : (1.0f / ((float)G_NB * (float)cnt));

  const int d4 = t & 31;   // float4 dim-group within a 128-float row

  // ---- phase A: edge gather (waves 0-3) + time encode (waves 4-7) ----
  if (t < 128) {
    const int kset = t >> 5;           // this wave's K-subset: k in [kset*8, kset*8+8)
    float4 acc = make_float4(0.f, 0.f, 0.f, 0.f);
    #pragma unroll
    for (int kk = 0; kk < 8; ++kk) {
      const int k = kset * 8 + kk;
      const float wk = s_w[k];
      const float4 e =
          ((const float4*)(edge_feat + (long long)s_eid[k] * EDIM))[d4];
      acc.x = fmaf(wk, e.x, acc.x);
      acc.y = fmaf(wk, e.y, acc.y);
      acc.z = fmaf(wk, e.z, acc.z);
      acc.w = fmaf(wk, e.w, acc.w);
    }
    s_red[t] = acc;
  } else if (t < CDIM) {
    const int d   = t - EDIM;
    const float w  = time_w[d];
    const float bb = time_b[d];
    float acc = 0.f;
    #pragma unroll 4
    for (int k = 0; k < K_NB; ++k) {
      if (s_valid[k]) acc += s_w[k] * cosf(fmaf(s_dt[k], w, bb));
    }
    v[(long long)b * VLD + t] = acc;
  }
  __syncthreads();
  if (t < 32) {
    float4 a0 = s_red[t], a1 = s_red[t + 32], a2 = s_red[t + 64], a3 = s_red[t + 96];
    float4 r;
    r.x = (a0.x + a1.x) + (a2.x + a3.x);
    r.y = (a0.y + a1.y) + (a2.y + a3.y);
    r.z = (a0.z + a1.z) + (a2.z + a3.z);
    r.w = (a0.w + a1.w) + (a2.w + a3.w);
    ((float4*)(v + (long long)b * VLD))[t] = r;
  }
  __syncthreads();   // s_red is reused below

  // ---- phase B: node gather, wave w owns rows g == w (mod 8) ----
  {
    const int rowset = t >> 5;
    float4 acc = make_float4(0.f, 0.f, 0.f, 0.f);
    for (int g = rowset; g < G_NB; g += 8) {
      if (d4 == 0) {   // one prefetch per wave, 32 rows ahead (global_prefetch_b8)
        const int gp = g + 32;
        if (gp < G_NB)
          __builtin_prefetch(node_feat + (long long)s_gid[gp] * NDIM, 0, 0);
      }
      const int id = s_gid[g];
      if (id > 0 || useAll) {
        const float4 x = ((const float4*)(node_feat + (long long)id * NDIM))[d4];
        acc.x += x.x; acc.y += x.y; acc.z += x.z; acc.w += x.w;
      }
    }
    s_red[t] = acc;
    __syncthreads();
    if (t < 32) {
      float4 r = make_float4(0.f, 0.f, 0.f, 0.f);
      #pragma unroll
      for (int rset = 0; rset < 8; ++rset) {
        const float4 a = s_red[t + 32 * rset];
        r.x += a.x; r.y += a.y; r.z += a.z; r.w += a.w;
      }
      const float4 self =
          ((const float4*)(node_feat + (long long)node_ids[b] * NDIM))[t];
      float4 o;
      o.x = fmaf(r.x, scale, self.x);
      o.y = fmaf(r.y, scale, self.y);
      o.z = fmaf(r.z, scale, self.z);
      o.w = fmaf(r.w, scale, self.w);
      ((float4*)(x2 + (long long)b * 256 + NDIM))[t] = o;
    }
  }
}

// ---------------------------------------------------------------------------
// f32 WMMA GEMM: out[M x N] = A[M x Kdim] @ W[Kdim x N] + bias
// One wave computes one 16x16 tile with V_WMMA_F32_16X16X4_F32.
// A layout per lane: rows row0+(lane&15), K slots k+2*(lane>>4), +1
// B layout per lane: cols col0+(lane&15), K slots k+2*(lane>>4), +1
// ---------------------------------------------------------------------------
__global__ __launch_bounds__(128)
void gemm_wmma_kernel(const float* __restrict__ A, int lda,
                      const float* __restrict__ W, int ldw,
                      int Kdim,
                      const float* __restrict__ bias,
                      float* __restrict__ out, int ldo,
                      int tilesN) {
  const int wave  = (blockIdx.x * blockDim.x + threadIdx.x) >> 5;
  const int lane  = threadIdx.x & 31;
  const int tileM = wave / tilesN;
  const int tileN = wave % tilesN;
  const int row0  = tileM * 16;
  const int col0  = tileN * 16;
  const int hl    = lane & 15;   // 0..15
  const int hi    = lane >> 4;   // 0 or 1

  const float* Arow = A + (long long)(row0 + hl) * lda;
  const float* Wcol = W + (col0 + hl);

  v8f c = {};
  for (int k = 0; k < Kdim; k += 4) {
    const int ka = k + 2 * hi;
    const v2f a = *(const v2f*)(Arow + ka);   // b64 load: 2 contiguous K slots
    v2f bfrag;
    bfrag.x = Wcol[(long long)ka * ldw];
    bfrag.y = Wcol[(long long)(ka + 1) * ldw];
    c = __builtin_amdgcn_wmma_f32_16x16x4_f32(
        /*neg_a=*/false, a, /*neg_b=*/false, bfrag,
        /*c_mod=*/(short)0, c, /*reuse_a=*/false, /*reuse_b=*/false);
  }

  const float bval = bias[col0 + hl];
  const int rbase = row0 + hi * 8;
  #pragma unroll
  for (int j = 0; j < 8; ++j) {
    out[(long long)(rbase + j) * ldo + col0 + hl] = c[j] + bval;
  }
}

// ---------------------------------------------------------------------------
extern "C" void kernel_launch(void* const* d_in, const int* in_sizes, int n_in,
                              void* d_out, int out_size, void* d_ws, size_t ws_size,
                              hipStream_t stream) {
  const int*   node_ids  = (const int*)  d_in[0];
  const float* t0        = (const float*)d_in[1];
  const int*   nbr_ids   = (const int*)  d_in[2];
  const int*   nbr_eids  = (const int*)  d_in[3];
  const float* nbr_times = (const float*)d_in[4];
  const int*   tg_ids    = (const int*)  d_in[5];
  const float* node_feat = (const float*)d_in[6];
  const float* edge_feat = (const float*)d_in[7];
  const float* time_w    = (const float*)d_in[8];
  const float* time_b    = (const float*)d_in[9];
  const float* W_edge    = (const float*)d_in[10];
  const float* b_edge    = (const float*)d_in[11];
  const float* w_final   = (const float*)d_in[12];
  const float* b_final   = (const float*)d_in[13];
  const float* W_node    = (const float*)d_in[14];
  const float* b_node    = (const float*)d_in[15];
  float* out = (float*)d_out;

  char*  ws          = (char*)d_ws;
  float* bias_pooled = (float*)ws;                                   // 128 f32
  float* v           = (float*)(ws + 1024);                          // 1024*232 f32
  float* x2          = (float*)(ws + 1024 + (size_t)B_SZ * VLD * 4); // 1024*256 f32

  prep_bias_kernel<<<1, 128, 0, stream>>>(w_final, b_edge, b_final, bias_pooled);

  gather_kernel<<<B_SZ, 256, 0, stream>>>(node_ids, t0, nbr_ids, nbr_eids,
                                          nbr_times, tg_ids, node_feat, edge_feat,
                                          time_w, time_b, w_final, v, x2);

  // pooled = v[1024x228] @ W_edge[228x128] + bias_pooled  ->  x2[:, 0:128]
  gemm_wmma_kernel<<<128, 128, 0, stream>>>(v, VLD, W_edge, EDIM, CDIM,
                                            bias_pooled, x2, 256, /*tilesN=*/8);

  // emb = x2[1024x256] @ W_node[256x128] + b_node  ->  d_out[1024x128]
  gemm_wmma_kernel<<<128, 128, 0, stream>>>(x2, 256, W_node, NDIM, 256,
                                            b_node, out, NDIM, /*tilesN=*/8);
}